// PaoReadout_747324309983
// MI455X (gfx1250) — compile-verified
//
#include <hip/hip_runtime.h>
#include <hip/hip_bf16.h>

typedef __attribute__((ext_vector_type(2))) float v2f;
typedef __attribute__((ext_vector_type(8))) float v8f;

#define TPB      128
#define ESTRIDE  343            // 169 (M / Asel+diag overlay) + 169 (U / vec overlay) + pad; gcd(343,64)=1
#define NSWEEP   10
#define CONSTF   2527           // 1407 wigner + 256 D (16x16 transposed, padded) + 864 W
#define SMEMF    (CONSTF + TPB * ESTRIDE)

// ---- slot tables (precomputed from the reference SLOTS enumeration) ----
__constant__ int S_DI[29]  = {1,1,1,3,3,3,3,3,3,3,3,3,3,3,3,3,5,5,5,5,5,5,5,5,5,5,5,5,5};
__constant__ int S_DJ[29]  = {1,1,1,1,1,3,3,3,1,1,3,3,3,3,3,3,1,1,3,3,3,3,3,3,5,5,5,5,5};
__constant__ int S_DK[29]  = {1,1,1,3,3,1,3,5,3,3,1,3,5,1,3,5,5,5,3,5,7,3,5,7,1,3,5,7,9};
__constant__ int S_A[29]   = {0,1,1,2,2,2,2,2,5,5,5,5,5,5,5,5,8,8,8,8,8,8,8,8,8,8,8,8,8};
__constant__ int S_B[29]   = {0,0,1,0,1,2,2,2,0,1,2,2,2,5,5,5,0,1,2,2,2,5,5,5,8,8,8,8,8};
__constant__ int S_WOFF[29]= {0,1,2,3,12,21,30,57,102,111,120,129,156,201,210,237,282,307,
                              332,377,452,557,602,677,782,807,882,1007,1182};
__constant__ int S_VOFF[29]= {0,1,2,3,6,9,10,13,18,21,24,25,28,33,34,37,42,47,52,55,60,67,
                              70,75,82,83,86,91,98};

// vec offsets per (lk,p) output group, channel order = SLOTS order
__constant__ int G0VOFF[7] = {0,1,2,9,24,33,82};     // (0,+)
__constant__ int G1VOFF[6] = {3,6,18,21,52,67};      // (1,-)
__constant__ int G2VOFF[4] = {10,25,34,83};          // (1,+)
__constant__ int G3VOFF[6] = {13,28,37,42,47,86};    // (2,+)
__constant__ int G4VOFF[2] = {55,70};                // (2,-)
__constant__ int G5VOFF[2] = {60,75};                // (3,-)
__constant__ int G6VOFF[1] = {91};                   // (3,+)
__constant__ int G7VOFF[1] = {98};                   // (4,+)

// equivariant linear for one irrep group: out[v,i] = (1/sqrt(MUL)) * sum_u x[u*DIM+i] * W[u*CNT+v]
template<int MUL, int CNT, int DIM>
__device__ inline void eqlin(const float* __restrict__ x, const float* __restrict__ W,
                             float scale, float* __restrict__ vecL, const int* __restrict__ voff)
{
    float acc[CNT * DIM];
#pragma unroll
    for (int i = 0; i < CNT * DIM; ++i) acc[i] = 0.0f;
    for (int u = 0; u < MUL; ++u) {
        float xu[DIM];
#pragma unroll
        for (int i = 0; i < DIM; ++i) xu[i] = x[u * DIM + i];
#pragma unroll
        for (int v = 0; v < CNT; ++v) {
            const float w = W[u * CNT + v];
#pragma unroll
            for (int i = 0; i < DIM; ++i) acc[v * DIM + i] = fmaf(xu[i], w, acc[v * DIM + i]);
        }
    }
#pragma unroll
    for (int v = 0; v < CNT; ++v)
#pragma unroll
        for (int i = 0; i < DIM; ++i)
            vecL[voff[v] + i] = acc[v * DIM + i] * scale;
}

__global__ __launch_bounds__(TPB) void pao_readout_kernel(
    const float* __restrict__ nodef, const int* __restrict__ cats,
    const float* __restrict__ W0e, const float* __restrict__ W1o,
    const float* __restrict__ W1e, const float* __restrict__ W2e,
    const float* __restrict__ W2o, const float* __restrict__ W3o,
    const float* __restrict__ W3e, const float* __restrict__ W4e,
    const float* __restrict__ wig, const float* __restrict__ Dm,
    float* __restrict__ out)
{
    extern __shared__ float smem[];
    float* wigL  = smem;                 // 1407
    float* DPT   = smem + 1407;          // 256: D transposed + zero-padded, DPT[n*16+k] = D[k][n]
    float* WL    = smem + 1663;          // 864 (all W concatenated)
    float* edges = smem + CONSTF;        // TPB * ESTRIDE

    const int tid = threadIdx.x;

    // ---------------- stage constants into LDS ----------------
    for (int i = tid; i < 1407; i += TPB) wigL[i] = wig[i];
    for (int i = tid; i < 256;  i += TPB) {
        const int n = i >> 4, k = i & 15;
        DPT[i] = (k < 13 && n < 13) ? Dm[k * 13 + n] : 0.0f;
    }
    {
        const float* wsrc[8] = {W0e, W1o, W1e, W2e, W2o, W3o, W3e, W4e};
        const int    woff[8] = {0, 224, 416, 544, 736, 800, 832, 848};
        const int    wsz [8] = {224, 192, 128, 192, 64, 32, 16, 16};
#pragma unroll
        for (int g = 0; g < 8; ++g)
            for (int i = tid; i < wsz[g]; i += TPB) WL[woff[g] + i] = wsrc[g][i];
    }
    __syncthreads();

    // ---------------- phase 1: one lane = one edge ----------------
    const int e    = blockIdx.x * TPB + tid;
    const int node = cats[e];
    const float* x = nodef + (size_t)node * 1088;

    float* Me   = edges + tid * ESTRIDE;     // 13x13 matrix (later: Asel[64] + diag[13] overlay)
    float* Ue   = Me + 169;                  // eigenvectors (overlaid with vec first)
    float* vecL = Ue;                        // 107-float TP coefficient vector

    const float r32 = 0.17677669529663687f;  // 1/sqrt(32)
    const float r16 = 0.25f;                 // 1/sqrt(16)
    eqlin<32, 7, 1>(x +   0, WL +   0, r32, vecL, G0VOFF);
    eqlin<32, 6, 3>(x + 160, WL + 224, r32, vecL, G1VOFF);
    eqlin<32, 4, 3>(x +  64, WL + 416, r32, vecL, G2VOFF);
    eqlin<32, 6, 5>(x + 256, WL + 544, r32, vecL, G3VOFF);
    eqlin<32, 2, 5>(x + 416, WL + 736, r32, vecL, G4VOFF);
    eqlin<16, 2, 7>(x + 688, WL + 800, r16, vecL, G5VOFF);
    eqlin<16, 1, 7>(x + 576, WL + 832, r16, vecL, G6VOFF);
    eqlin<16, 1, 9>(x + 800, WL + 848, r16, vecL, G7VOFF);

    // ---- M = I + sum_s wigner[s] . vec[s] (lower/diag blocks), then symmetrize ----
    for (int i = 0; i < 169; ++i) Me[i] = 0.0f;
    for (int i = 0; i < 13;  ++i) Me[i * 13 + i] = 1.0f;
    for (int s = 0; s < 29; ++s) {
        const int di = S_DI[s], dj = S_DJ[s], dk = S_DK[s];
        const int a = S_A[s], b = S_B[s];
        const float* w  = wigL + S_WOFF[s];
        const float* vv = vecL + S_VOFF[s];
        for (int i = 0; i < di; ++i)
            for (int j = 0; j < dj; ++j) {
                float acc = 0.0f;
                const float* wp = w + (i * dj + j) * dk;
                for (int k = 0; k < dk; ++k) acc = fmaf(vv[k], wp[k], acc);
                Me[(a + i) * 13 + (b + j)] += acc;
            }
    }
    // Msym = tril(M) + tril(M,-1)^T  ->  mirror lower into upper
    for (int i = 0; i < 13; ++i)
        for (int j = i + 1; j < 13; ++j)
            Me[i * 13 + j] = Me[j * 13 + i];

    // ---- cyclic Jacobi eigensolver; U accumulates rotations ----
    for (int i = 0; i < 169; ++i) Ue[i] = 0.0f;   // vec no longer needed
    for (int i = 0; i < 13;  ++i) Ue[i * 13 + i] = 1.0f;

    for (int sweep = 0; sweep < NSWEEP; ++sweep) {
        for (int p = 0; p < 12; ++p) {
            for (int q = p + 1; q < 13; ++q) {
                const float apq = Me[p * 13 + q];
                if (fabsf(apq) > 1e-12f) {
                    const float app = Me[p * 13 + p];
                    const float aqq = Me[q * 13 + q];
                    const float tau = (aqq - app) / (2.0f * apq);
                    const float t   = (tau >= 0.0f ? 1.0f : -1.0f) /
                                      (fabsf(tau) + sqrtf(1.0f + tau * tau));
                    const float c = rsqrtf(1.0f + t * t);
                    const float s = t * c;
                    for (int k = 0; k < 13; ++k) {        // A <- A*G  (columns p,q)
                        const float akp = Me[k * 13 + p], akq = Me[k * 13 + q];
                        Me[k * 13 + p] = c * akp - s * akq;
                        Me[k * 13 + q] = s * akp + c * akq;
                    }
                    for (int k = 0; k < 13; ++k) {        // A <- G^T*A (rows p,q)
                        const float apk = Me[p * 13 + k], aqk = Me[q * 13 + k];
                        Me[p * 13 + k] = c * apk - s * aqk;
                        Me[q * 13 + k] = s * apk + c * aqk;
                    }
                    for (int k = 0; k < 13; ++k) {        // U <- U*G
                        const float ukp = Ue[k * 13 + p], ukq = Ue[k * 13 + q];
                        Ue[k * 13 + p] = c * ukp - s * ukq;
                        Ue[k * 13 + q] = s * ukp + c * ukq;
                    }
                }
            }
        }
    }

    // ---- select 4 smallest eigenvalues (ascending); stage zero-padded A tile ----
    // Copy diagonal to Me[64..76] (safe: only src overlapping dest is Me[70], read before written).
    for (int i = 0; i < 13; ++i) Me[64 + i] = Me[i * 13 + i];
    {
        unsigned used = 0u;
        for (int r = 0; r < 4; ++r) {
            float best = 3.0e38f; int bi = 0;
            for (int i = 0; i < 13; ++i) {
                if (!((used >> i) & 1u)) {
                    const float v = Me[64 + i];
                    if (v < best) { best = v; bi = i; }
                }
            }
            used |= 1u << bi;
            // Asel[r*16 + k] = U[k][bi], zero-padded to K=16  (overlays Me[0..63])
            for (int k = 0; k < 13; ++k) Me[r * 16 + k] = Ue[k * 13 + bi];
            Me[r * 16 + 13] = 0.0f;
            Me[r * 16 + 14] = 0.0f;
            Me[r * 16 + 15] = 0.0f;
        }
    }
    __syncthreads();

    // ---------------- phase 2: WMMA output transform ----------------
    // out[e, r, j] = sum_k Asel_e[r][k] * D[k][j]
    // 4 edges packed -> A(16x16, pre-padded), B = DPT (pre-padded); 4x chained 16x16x4 f32 WMMA.
    // All fragment loads are unconditional: EXEC stays all-1s through the WMMA chain.
    const int lane = tid & 31;
    const int wave = tid >> 5;
    const int m16  = lane & 15;
    const int half = lane >> 4;

#pragma unroll 1
    for (int g = 0; g < 8; ++g) {
        const int elA = g * 4 + (m16 >> 2);      // which edge this A-row belongs to
        const int rA  = m16 & 3;                 // which PAO row
        const float* Aa = edges + (wave * 32 + elA) * ESTRIDE + rA * 16;

        v8f acc = {};
#pragma unroll
        for (int k0 = 0; k0 < 16; k0 += 4) {
            const int ka = k0 + half * 2;        // this lane's two K slots
            v2f a, b;
            a[0] = Aa[ka];
            a[1] = Aa[ka + 1];
            b[0] = DPT[m16 * 16 + ka];
            b[1] = DPT[m16 * 16 + ka + 1];
            acc = __builtin_amdgcn_wmma_f32_16x16x4_f32(
                false, a, false, b, (short)0, acc, false, false);
        }
        if (m16 < 13) {
#pragma unroll
            for (int v = 0; v < 8; ++v) {
                const int Mrow = v + half * 8;   // D-matrix row this VGPR holds
                const int el   = g * 4 + (Mrow >> 2);
                const int r    = Mrow & 3;
                const size_t eg = (size_t)blockIdx.x * TPB + wave * 32 + el;
                out[eg * 52 + r * 13 + m16] = acc[v];
            }
        }
    }
}

extern "C" void kernel_launch(void* const* d_in, const int* in_sizes, int n_in,
                              void* d_out, int out_size, void* d_ws, size_t ws_size,
                              hipStream_t stream)
{
    const float* nodef = (const float*)d_in[0];
    const int*   cats  = (const int*)  d_in[1];
    const float* W0e   = (const float*)d_in[2];
    const float* W1o   = (const float*)d_in[3];
    const float* W1e   = (const float*)d_in[4];
    const float* W2e   = (const float*)d_in[5];
    const float* W2o   = (const float*)d_in[6];
    const float* W3o   = (const float*)d_in[7];
    const float* W3e   = (const float*)d_in[8];
    const float* W4e   = (const float*)d_in[9];
    const float* wig   = (const float*)d_in[10];
    const float* Dm    = (const float*)d_in[11];
    float*       out   = (float*)d_out;

    const int nE = in_sizes[1];                  // number of edges
    const int smemBytes = SMEMF * (int)sizeof(float);   // ~181 KB (<320 KB/WGP)

    hipFuncSetAttribute((const void*)pao_readout_kernel,
                        hipFuncAttributeMaxDynamicSharedMemorySize, smemBytes);

    dim3 grid(nE / TPB), block(TPB);
    pao_readout_kernel<<<grid, block, smemBytes, stream>>>(
        nodef, cats, W0e, W1o, W1e, W2e, W2o, W3o, W3e, W4e, wig, Dm, out);
}